// GConv_71485435674712
// MI455X (gfx1250) — compile-verified
//
#include <hip/hip_runtime.h>
#include <hip/hip_bf16.h>

typedef float v2f __attribute__((ext_vector_type(2)));
typedef float v8f __attribute__((ext_vector_type(8)));

#define FDIM 64
#define LDSTRIDE 68   // 68 mod 64 = 4 -> consecutive lanes hit distinct bank pairs on ds_load_b64

// ---------------------------------------------------------------- zero buffer
__global__ void GConv_zero_kernel(float4* __restrict__ p, int n4) {
    int i = blockIdx.x * blockDim.x + threadIdx.x;
    if (i < n4) p[i] = make_float4(0.f, 0.f, 0.f, 0.f);
}

// ------------------------------------------------------------- scatter (E·F atomics)
// 16 threads per edge; each does a float4 gather from x[src] and 4 atomic adds to agg[dst].
__global__ void GConv_scatter_kernel(const float* __restrict__ x,
                                     const int* __restrict__ ei,   // [2,E] flat: src row then dst row
                                     float* __restrict__ agg, int E) {
    int t = blockIdx.x * blockDim.x + threadIdx.x;
    int edge = t >> 4;
    if (edge >= E) return;
    int f = (t & 15) * 4;
    int src = ei[edge];
    int dst = ei[E + edge];
    const float4 v = *(const float4*)(x + (size_t)src * FDIM + f);
    float* p = agg + (size_t)dst * FDIM + f;
    atomicAdd(p + 0, v.x);
    atomicAdd(p + 1, v.y);
    atomicAdd(p + 2, v.z);
    atomicAdd(p + 3, v.w);
}

// --------------------------------------------- fused dual-GEMM + ReLU via fp32 WMMA
// xout[m,n] = relu( sum_k agg[m,k]*Wrel[n,k] + sum_k xin[m,k]*Wroot[n,k] )
// One wave per 16-row tile; 8 waves (256 thr) per block; W matrices staged in LDS.
__global__ __launch_bounds__(256) void GConv_gemm_relu_kernel(
        const float* __restrict__ agg, const float* __restrict__ xin,
        const float* __restrict__ Wrel, const float* __restrict__ Wroot,
        float* __restrict__ xout, int nrows) {
    __shared__ float sWrel[FDIM * LDSTRIDE];
    __shared__ float sWroot[FDIM * LDSTRIDE];

    const int t = threadIdx.x;
    // Cooperative stage of both 64x64 weight matrices into LDS (padded rows).
    for (int i = t; i < FDIM * FDIM; i += 256) {
        int r = i >> 6, c = i & 63;
        sWrel[r * LDSTRIDE + c]  = Wrel[i];
        sWroot[r * LDSTRIDE + c] = Wroot[i];
    }
    __syncthreads();

    const int wave = t >> 5;
    const int lane = t & 31;
    const int tile = blockIdx.x * 8 + wave;
    const int m0 = tile * 16;
    if (m0 >= nrows) return;                 // wave-uniform exit: EXEC stays all-ones for WMMA

    const int mrow = m0 + (lane & 15);       // A-fragment row for this lane
    const int koff = (lane >> 4) * 2;        // lanes 0-15: K={0,1}; lanes 16-31: K={2,3}
    const int ncol = lane & 15;              // B-fragment column for this lane

    // Preload all 16 K-step A-fragments for both operand matrices (in-place safe).
    v2f a_agg[16], a_x[16];
#pragma unroll
    for (int k = 0; k < 16; ++k) {
        const int kb = k * 4 + koff;
        a_agg[k] = *(const v2f*)(agg + (size_t)mrow * FDIM + kb);
        a_x[k]   = *(const v2f*)(xin + (size_t)mrow * FDIM + kb);
    }

    v8f acc[4] = {};                         // 16x64 output row-block, 4 N-tiles
#pragma unroll
    for (int nt = 0; nt < 4; ++nt) {
        const float* br_base = sWrel  + (size_t)(nt * 16 + ncol) * LDSTRIDE;
        const float* bo_base = sWroot + (size_t)(nt * 16 + ncol) * LDSTRIDE;
#pragma unroll
        for (int k = 0; k < 16; ++k) {
            const int kb = k * 4 + koff;
            v2f br = *(const v2f*)(br_base + kb);   // B[k][n] = Wrel[n][k]
            acc[nt] = __builtin_amdgcn_wmma_f32_16x16x4_f32(
                false, a_agg[k], false, br, (short)0, acc[nt], false, false);
            v2f bo = *(const v2f*)(bo_base + kb);   // B[k][n] = Wroot[n][k]
            acc[nt] = __builtin_amdgcn_wmma_f32_16x16x4_f32(
                false, a_x[k], false, bo, (short)0, acc[nt], false, false);
        }
    }

    // ReLU + store. C/D layout: VGPR r -> row m0+r (lanes 0-15) / m0+r+8 (lanes 16-31), lane&15 -> col.
    const int rbase = m0 + (lane >> 4) * 8;
#pragma unroll
    for (int nt = 0; nt < 4; ++nt) {
        const int c = nt * 16 + (lane & 15);
#pragma unroll
        for (int r = 0; r < 8; ++r) {
            float v = acc[nt][r];
            v = v > 0.f ? v : 0.f;
            xout[(size_t)(rbase + r) * FDIM + c] = v;
        }
    }
}

// ---------------------------------------------------------------- readout x @ e.T
// One wave per node: lane handles 2 features, wave32 xor-shuffle reduction.
__global__ void GConv_readout_kernel(const float* __restrict__ x,
                                     const float* __restrict__ e,
                                     float* __restrict__ out, int n) {
    int g = blockIdx.x * blockDim.x + threadIdx.x;
    int node = g >> 5;
    int lane = threadIdx.x & 31;
    if (node >= n) return;
    const float2 xv = *(const float2*)(x + (size_t)node * FDIM + lane * 2);
    const float2 ev = *(const float2*)(e + lane * 2);
    float v = xv.x * ev.x + xv.y * ev.y;
#pragma unroll
    for (int off = 16; off > 0; off >>= 1) v += __shfl_xor(v, off, 32);
    if (lane == 0) out[node] = v;
}

extern "C" void kernel_launch(void* const* d_in, const int* in_sizes, int n_in,
                              void* d_out, int out_size, void* d_ws, size_t ws_size,
                              hipStream_t stream) {
    const float* x     = (const float*)d_in[0];
    const int*   ei    = (const int*)d_in[1];   // [2,E]
    // d_in[2] = batch (unused by forward)
    const float* Wrel  = (const float*)d_in[3];
    const float* Wroot = (const float*)d_in[4];
    const float* evec  = (const float*)d_in[5];
    float* out = (float*)d_out;

    const int N = in_sizes[0] / FDIM;   // 50000
    const int E = in_sizes[1] / 2;      // 800000

    float* agg = (float*)d_ws;                  // [N,64]
    float* xa  = agg + (size_t)N * FDIM;        // [N,64] ping buffer (gemm is in-place safe)

    const int n4 = (N * FDIM) / 4;
    const int zgrid = (n4 + 255) / 256;
    const int sgrid = (E * 16 + 255) / 256;
    const int tiles = (N + 15) / 16;
    const int ggrid = (tiles + 7) / 8;
    const int rgrid = (N * 32 + 255) / 256;

    // iteration 0: read from input x, write xa
    GConv_zero_kernel<<<zgrid, 256, 0, stream>>>((float4*)agg, n4);
    GConv_scatter_kernel<<<sgrid, 256, 0, stream>>>(x, ei, agg, E);
    GConv_gemm_relu_kernel<<<ggrid, 256, 0, stream>>>(agg, x, Wrel, Wroot, xa, N);

    // iterations 1..2: xa -> xa (row-block in-place)
    for (int it = 1; it < 3; ++it) {
        GConv_zero_kernel<<<zgrid, 256, 0, stream>>>((float4*)agg, n4);
        GConv_scatter_kernel<<<sgrid, 256, 0, stream>>>(xa, ei, agg, E);
        GConv_gemm_relu_kernel<<<ggrid, 256, 0, stream>>>(agg, xa, Wrel, Wroot, xa, N);
    }

    GConv_readout_kernel<<<rgrid, 256, 0, stream>>>(xa, evec, out, N);
}